// APCriterion_68899865362854
// MI455X (gfx1250) — compile-verified
//
#include <hip/hip_runtime.h>

// APCriterion on MI455X (gfx1250, wave32).
// sim = anc(8192x128) @ pos^T via V_WMMA_F32_16X16X4_F32 (exact f32), fused
// with per-row streaming top-20 over non-same-crop columns; then 20-bin
// quantized AP per row; deterministic two-stage mean reduction.

#define N_CROPS 64
#define KCROP   128
#define DDIM    128
#define BTOT    (N_CROPS * KCROP)   // 8192
#define KNN     20
#define NQ      20
#define NBLK    (BTOT / 16)         // 512 row blocks
#define NGRP    (BTOT / 64)         // 128 column groups (4 tiles of 16 each)

typedef __attribute__((ext_vector_type(2))) float v2f;
typedef __attribute__((ext_vector_type(8))) float v8f;

// Soft-quantization kernel bins (triangular partition over [0,1], nq=20):
//   k_i(x) = max(0, min(w1_i*x + b1_i, w2_i*x + b2_i))
// w1 = -19 (i==0: 0), b1 = 20-i (i==0: 1); w2 = 19 (i==19: 0), b2 = i-18.
__device__ __forceinline__ void accum_bins(float x, float* bins) {
#pragma unroll
  for (int i = 0; i < NQ; ++i) {
    const float w1 = (i == 0)      ? 0.0f : -19.0f;
    const float b1 = (i == 0)      ? 1.0f : (float)(20 - i);
    const float w2 = (i == NQ - 1) ? 0.0f : 19.0f;
    const float b2 = (float)(i - 18);   // i==19 gives 1.0 (matches override)
    float q = fminf(fmaf(w1, x, b1), fmaf(w2, x, b2));
    bins[i] += fmaxf(q, 0.0f);
  }
}

__global__ __launch_bounds__(32)
void ap_main(const float* __restrict__ anc, const float* __restrict__ pos,
             float* __restrict__ ws) {
  __shared__ float tileb[4][16][17];      // padded: conflict-free row reads
  __shared__ float topb[32][KNN];         // per half-lane unsorted top-20
  __shared__ float pospart[32];
  __shared__ float apb[16];

  const int lane = threadIdx.x;           // 0..31
  const int r    = lane & 15;             // row within block / col within tile
  const int hi   = lane >> 4;             // half-wave id
  const int row0 = blockIdx.x * 16;
  const int crop = blockIdx.x >> 3;       // 8 row-blocks per crop

  // ---- init per-half-lane top-20 (unit-norm sims are in [-1,1]) ----
#pragma unroll
  for (int j = 0; j < KNN; ++j) topb[lane][j] = -3.0f;
  float curmin = -3.0f;
  int   minidx = 0;

  // ---- A fragment, resident in registers for the whole kernel ----
  // V_WMMA_F32_16X16X4_F32 A layout: lane m (0-15) K={0,1}, lane m+16 K={2,3}
  v2f afrag[32];
  {
    const float* arow = anc + (size_t)(row0 + r) * DDIM + 2 * hi;
#pragma unroll
    for (int kk = 0; kk < 32; ++kk)
      afrag[kk] = *(const v2f*)(arow + 4 * kk);
  }

  // ---- sim_pos[r] = dot(anc[row], pos[row]) : split across half-waves ----
  {
    const float* ar = anc + (size_t)(row0 + r) * DDIM + hi * 64;
    const float* pr = pos + (size_t)(row0 + r) * DDIM + hi * 64;
    float s = 0.0f;
#pragma unroll 8
    for (int j = 0; j < 64; ++j) s = fmaf(ar[j], pr[j], s);
    pospart[lane] = s;
  }
  __syncthreads();
  const float simpos = pospart[r] + pospart[r + 16];

  // ---- main loop: 128 groups of 4 column tiles (64 cols each) ----
  const int exg0 = crop * 2;              // 2 excluded groups = same crop
#pragma unroll 1
  for (int g = 0; g < NGRP; ++g) {
    if (g == exg0 || g == exg0 + 1) continue;
    const int colbase = g * 64;

    v8f acc[4];
#pragma unroll
    for (int t = 0; t < 4; ++t) {
      v8f z = {0.f, 0.f, 0.f, 0.f, 0.f, 0.f, 0.f, 0.f};
      acc[t] = z;
    }

    // B layout: lane n (0-15) holds col n, K={k0,k0+1}; lane n+16 K={k0+2,k0+3}
    const float* bp[4];
#pragma unroll
    for (int t = 0; t < 4; ++t)
      bp[t] = pos + (size_t)(colbase + t * 16 + r) * DDIM + 2 * hi;

#pragma unroll
    for (int kk = 0; kk < 32; ++kk) {
#pragma unroll
      for (int t = 0; t < 4; ++t) {
        v2f bfrag = *(const v2f*)(bp[t] + 4 * kk);
        acc[t] = __builtin_amdgcn_wmma_f32_16x16x4_f32(
            false, afrag[kk], false, bfrag, (short)0, acc[t], false, false);
      }
    }

    // C/D layout: VGPR v holds row (v + 8*hi), lane selects col r.
    __syncthreads();   // WAR vs previous group's candidate reads
#pragma unroll
    for (int t = 0; t < 4; ++t)
#pragma unroll
      for (int v = 0; v < 8; ++v)
        tileb[t][v + 8 * hi][r] = acc[t][v];
    __syncthreads();

    // streaming top-20: lane owns row r, half hi scans 8 cols per tile
#pragma unroll 1
    for (int t = 0; t < 4; ++t) {
#pragma unroll
      for (int j = 0; j < 8; ++j) {
        float cand = tileb[t][r][hi * 8 + j];
        if (cand > curmin) {
          topb[lane][minidx] = cand;
          curmin = topb[lane][0];
          minidx = 0;
#pragma unroll
          for (int q = 1; q < KNN; ++q) {
            float vq = topb[lane][q];
            if (vq < curmin) { curmin = vq; minidx = q; }
          }
        }
      }
    }
  }
  __syncthreads();

  // ---- per-row: merge two top-20s, bin, compute AP (lanes 0-15) ----
  if (lane < 16) {
    float nbs[NQ], rec[NQ];
#pragma unroll
    for (int i = 0; i < NQ; ++i) { nbs[i] = 0.0f; rec[i] = 0.0f; }

    accum_bins(simpos, rec);            // positive column (label row)

    // top-20 of the 40 merged candidates (AP is order-invariant over negs)
    for (int sel = 0; sel < KNN; ++sel) {
      float best = -1e30f; int bj = 0;
#pragma unroll
      for (int j = 0; j < 2 * KNN; ++j) {
        float v = (j < KNN) ? topb[r][j] : topb[r + 16][j - KNN];
        if (v > best) { best = v; bj = j; }
      }
      if (bj < KNN) topb[r][bj] = -1e30f; else topb[r + 16][bj - KNN] = -1e30f;
      accum_bins(best, nbs);
    }
#pragma unroll
    for (int i = 0; i < NQ; ++i) nbs[i] += rec[i];   // nbs sums all 21 cols

    float cr = 0.0f, cn = 0.0f, rsum = 0.0f, ap = 0.0f;
#pragma unroll
    for (int i = 0; i < NQ; ++i) {
      cr += rec[i];
      cn += nbs[i];
      ap += rec[i] * (cr / (1e-16f + cn));
      rsum += rec[i];
    }
    apb[r] = ap / rsum;
  }
  __syncthreads();

  if (lane == 0) {                      // deterministic fixed-order WG sum
    float s = 0.0f;
    for (int j = 0; j < 16; ++j) s += apb[j];
    ws[blockIdx.x] = s;
  }
}

__global__ void ap_finalize(const float* __restrict__ ws,
                            float* __restrict__ out) {
  if (threadIdx.x == 0) {
    float s = 0.0f;
    for (int i = 0; i < NBLK; ++i) s += ws[i];   // fixed order: deterministic
    float mean_ap = s / (float)BTOT;
    out[0] = 1.0f - mean_ap;
    out[1] = mean_ap;
  }
}

extern "C" void kernel_launch(void* const* d_in, const int* in_sizes, int n_in,
                              void* d_out, int out_size, void* d_ws, size_t ws_size,
                              hipStream_t stream) {
  const float* anc = (const float*)d_in[0];
  const float* pos = (const float*)d_in[1];
  // d_in[2] (kpts_crop_ids) is uniformly 128 per the reference assertion; unused.
  float* ws  = (float*)d_ws;
  float* out = (float*)d_out;

  ap_main<<<NBLK, 32, 0, stream>>>(anc, pos, ws);
  ap_finalize<<<1, 32, 0, stream>>>(ws, out);
}